// GADBase_8538394984912
// MI455X (gfx1250) — compile-verified
//
#include <hip/hip_runtime.h>
#include <stdint.h>

// Problem constants (match reference)
#define HH 1024
#define WW 1024
#define SHH 128
#define NPRE 64
#define LSTEP 0.24f
#define KKC (0.03f * 0.03f)
#define EPSC 1e-8f
#define DEPSC 0.1f

// ---------------------------------------------------------------------------
// CDNA5 async global->LDS helpers (gfx1250): tracked by ASYNCcnt.
// ---------------------------------------------------------------------------
__device__ __forceinline__ uint32_t lds_off_of(const void* p) {
  // AS3 -> generic addrspacecast puts the LDS byte offset in the low 32 bits
  // (SHARED aperture base has zero low bits), so truncation yields the DS addr.
  return (uint32_t)(uintptr_t)p;
}

__device__ __forceinline__ void async_g2l_b32(uint32_t lds_byte,
                                              const float* base,
                                              uint32_t gbyte_off) {
  // GVS form: global addr = SADDR(64) + VADDR(32) ; VDST holds LDS byte addr.
  asm volatile("global_load_async_to_lds_b32 %0, %1, %2"
               :
               : "v"(lds_byte), "v"(gbyte_off), "s"(base)
               : "memory");
}

__device__ __forceinline__ void wait_async0() {
#if __has_builtin(__builtin_amdgcn_s_wait_asynccnt)
  __builtin_amdgcn_s_wait_asynccnt(0);
#else
  asm volatile("s_wait_asynccnt 0" ::: "memory");
#endif
}

// ---------------------------------------------------------------------------
// 1) shift = (min(source) <= 0.1) ? 0.1 : 0
// ---------------------------------------------------------------------------
__global__ __launch_bounds__(256) void k_min(const float* __restrict__ src,
                                             float* __restrict__ shift_out) {
  __shared__ float red[8];
  const int t = threadIdx.x;
  float m = 3.4e38f;
  for (int i = t; i < 2 * SHH * SHH; i += 256) m = fminf(m, src[i]);
#pragma unroll
  for (int d = 16; d; d >>= 1) m = fminf(m, __shfl_xor(m, d, 32));
  if ((t & 31) == 0) red[t >> 5] = m;
  __syncthreads();
  if (t == 0) {
    float mm = red[0];
#pragma unroll
    for (int i = 1; i < 8; ++i) mm = fminf(mm, red[i]);
    shift_out[0] = (mm <= DEPSC) ? DEPSC : 0.0f;
  }
}

// ---------------------------------------------------------------------------
// 2) conductance maps cv [B,1023,1024], ch [B,1024,1023]
//    feats = concat(guide(3ch), img); img diffs == y diffs (shift cancels)
// ---------------------------------------------------------------------------
__global__ __launch_bounds__(256) void k_cond(const float* __restrict__ guide,
                                              const float* __restrict__ y,
                                              float* __restrict__ cv,
                                              float* __restrict__ ch) {
  const int idx = blockIdx.x * 256 + threadIdx.x;   // [0, 2*1024*1024)
  const int j = idx & (WW - 1);
  const int i = (idx >> 10) & (HH - 1);
  const int b = idx >> 20;
  const size_t HW = (size_t)HH * WW;
  const float* g0 = guide + (size_t)b * 3 * HW;
  const size_t p = (size_t)i * WW + j;
  const float yc = y[(size_t)b * HW + p];
  const float gc0 = g0[p], gc1 = g0[HW + p], gc2 = g0[2 * HW + p];
  if (i < HH - 1) {
    const size_t pd = p + WW;
    float s = fabsf(g0[pd] - gc0) + fabsf(g0[HW + pd] - gc1) +
              fabsf(g0[2 * HW + pd] - gc2) + fabsf(y[(size_t)b * HW + pd] - yc);
    s *= 0.25f;
    cv[(size_t)b * (HH - 1) * WW + p] = 1.0f / (1.0f + (s * s) / KKC);
  }
  if (j < WW - 1) {
    const size_t pr = p + 1;
    float s = fabsf(g0[pr] - gc0) + fabsf(g0[HW + pr] - gc1) +
              fabsf(g0[2 * HW + pr] - gc2) + fabsf(y[(size_t)b * HW + pr] - yc);
    s *= 0.25f;
    ch[(size_t)b * HH * (WW - 1) + (size_t)i * (WW - 1) + j] =
        1.0f / (1.0f + (s * s) / KKC);
  }
}

// ---------------------------------------------------------------------------
// 3) img0 = y + shift
// ---------------------------------------------------------------------------
__global__ __launch_bounds__(256) void k_init(const float* __restrict__ y,
                                              const float* __restrict__ shift,
                                              float* __restrict__ img) {
  const int i = blockIdx.x * 256 + threadIdx.x;
  img[i] = y[i] + shift[0];
}

// ---------------------------------------------------------------------------
// 4) one diffusion step fused with the 8x8 ratio adjust.
//    Tile = 64 wide x 8 high per workgroup (256 thr, 2 px/thread, wave32x8).
//    Image tile (+halo) AND cv/ch slabs all staged to LDS via the CDNA5
//    async global->LDS engine; one s_wait_asynccnt + barrier covers all.
// ---------------------------------------------------------------------------
__global__ __launch_bounds__(256) void k_step(const float* __restrict__ in,
                                              float* __restrict__ out,
                                              const float* __restrict__ cv,
                                              const float* __restrict__ ch,
                                              const float* __restrict__ src,
                                              const float* __restrict__ mask,
                                              const float* __restrict__ shift_p) {
  __shared__ float tile[10][72];   // img rows -1..8 -> [0..9], cols -1..64 -> [0..65]
  __shared__ float cvt[9][64];     // cv rows i0-1..i0+7, cols j0..j0+63
  __shared__ float cht[8][65];     // ch rows i0..i0+7,  cols j0-1..j0+63
  __shared__ float part[32];       // 8 waves x 4 groups
  __shared__ float ratio[8];       // one per 8x8 block in tile

  const int t = threadIdx.x;
  const int b = blockIdx.z;
  const int i0 = blockIdx.y * 8;
  const int j0 = blockIdx.x * 64;
  const size_t plane = (size_t)b * HH * WW;
  const int r = t >> 6;            // 0..3 (thread also handles row r+4)
  const int c = t & 63;

  // ---- async stage image interior 8x64 (two rows per thread) ----
  {
    const uint32_t go0 =
        (uint32_t)((plane + (size_t)(i0 + r) * WW + (j0 + c)) * 4u);
    async_g2l_b32(lds_off_of(&tile[r + 1][c + 1]), in, go0);
    async_g2l_b32(lds_off_of(&tile[r + 5][c + 1]), in, go0 + 4u * 4u * WW);
  }

  // ---- async stage image halo (clamped; true-edge values are never used:
  //      edge flux terms are predicated to zero below) ----
  if (t < 148) {
    int gi, gj, lr, lc;
    if (t < 66)       { gi = i0 - 1;       gj = j0 - 1 + t;  lr = 0;       lc = t;      }
    else if (t < 132) { gi = i0 + 8;       gj = j0 - 67 + t; lr = 9;       lc = t - 66; }
    else if (t < 140) { gi = i0 + t - 132; gj = j0 - 1;      lr = t - 131; lc = 0;      }
    else              { gi = i0 + t - 140; gj = j0 + 64;     lr = t - 139; lc = 65;     }
    const int ci = min(max(gi, 0), HH - 1);
    const int cj = min(max(gj, 0), WW - 1);
    async_g2l_b32(lds_off_of(&tile[lr][lc]), in,
                  (uint32_t)((plane + (size_t)ci * WW + cj) * 4u));
  }

  // ---- async stage cv slab: rows i0-1..i0+7 (clamped), cols j0..j0+63 ----
  {
    const size_t cvb = (size_t)b * (HH - 1) * WW;
    for (int idx = t; idx < 9 * 64; idx += 256) {
      const int row = idx >> 6, col = idx & 63;
      const int grow = min(max(i0 - 1 + row, 0), HH - 2);
      async_g2l_b32(lds_off_of(&cvt[row][col]), cv,
                    (uint32_t)((cvb + (size_t)grow * WW + (j0 + col)) * 4u));
    }
  }

  // ---- async stage ch slab: rows i0..i0+7, cols j0-1..j0+63 (clamped) ----
  {
    const size_t chb = (size_t)b * HH * (WW - 1);
    for (int idx = t; idx < 8 * 65; idx += 256) {
      const int row = idx / 65, col = idx - 65 * row;
      const int gcol = min(max(j0 - 1 + col, 0), WW - 2);
      async_g2l_b32(lds_off_of(&cht[row][col]), ch,
                    (uint32_t)((chb + (size_t)(i0 + row) * (WW - 1) + gcol) * 4u));
    }
  }

  wait_async0();
  __syncthreads();

  // ---- stencil (Jacobi: all diffs from the pre-step image), all from LDS ----
  float nv[2];
#pragma unroll
  for (int k = 0; k < 2; ++k) {
    const int rr = r + 4 * k;
    const int gi = i0 + rr, gj = j0 + c;
    const float imc = tile[rr + 1][c + 1];
    const float imu = tile[rr][c + 1];
    const float imd = tile[rr + 2][c + 1];
    const float iml = tile[rr + 1][c];
    const float imr = tile[rr + 1][c + 2];
    const float cvd = (gi < HH - 1) ? cvt[rr + 1][c] : 0.0f;   // cv[gi]
    const float cvu = (gi > 0) ? cvt[rr][c] : 0.0f;            // cv[gi-1]
    const float chr_ = (gj < WW - 1) ? cht[rr][c + 1] : 0.0f;  // ch[gj]
    const float chl = (gj > 0) ? cht[rr][c] : 0.0f;            // ch[gj-1]
    nv[k] = imc + LSTEP * (cvd * (imd - imc) - cvu * (imc - imu) +
                           chr_ * (imr - imc) - chl * (imc - iml));
  }

  // ---- 8x8 block mean: wave32 shuffle partials + LDS combine ----
  float s = nv[0] + nv[1];                     // both px share a column/block
  s += __shfl_xor(s, 1, 32);
  s += __shfl_xor(s, 2, 32);
  s += __shfl_xor(s, 4, 32);                   // 16-px partial per 8-lane group
  const int lane = t & 31, wave = t >> 5;
  if ((lane & 7) == 0) part[wave * 4 + (lane >> 3)] = s;
  __syncthreads();

  if (t < 8) {                                 // block t: cols 8t..8t+7
    const int g = t & 3, p0 = t >> 2;          // wave parity
    const float total = part[(p0 + 0) * 4 + g] + part[(p0 + 2) * 4 + g] +
                        part[(p0 + 4) * 4 + g] + part[(p0 + 6) * 4 + g];
    const float mean = total * (1.0f / 64.0f);
    const size_t sidx =
        ((size_t)b * SHH + blockIdx.y) * SHH + (blockIdx.x * 8 + t);
    const float sv = src[sidx] + shift_p[0];
    ratio[t] = (mask[sidx] < 0.5f) ? 1.0f : sv / (mean + EPSC);
  }
  __syncthreads();

  const float rt = ratio[c >> 3];
  const size_t o0 = plane + (size_t)(i0 + r) * WW + (j0 + c);
  out[o0] = nv[0] * rt;
  out[o0 + (size_t)4 * WW] = nv[1] * rt;
}

// ---------------------------------------------------------------------------
// 5) y_pred = img - shift
// ---------------------------------------------------------------------------
__global__ __launch_bounds__(256) void k_final(const float* __restrict__ img,
                                               const float* __restrict__ shift,
                                               float* __restrict__ yp) {
  const int i = blockIdx.x * 256 + threadIdx.x;
  yp[i] = img[i] - shift[0];
}

// ---------------------------------------------------------------------------
extern "C" void kernel_launch(void* const* d_in, const int* in_sizes, int n_in,
                              void* d_out, int out_size, void* d_ws,
                              size_t ws_size, hipStream_t stream) {
  (void)in_sizes; (void)n_in; (void)out_size; (void)ws_size;
  const float* guide = (const float*)d_in[0];   // [2,3,1024,1024]
  const float* ybic  = (const float*)d_in[1];   // [2,1,1024,1024]
  const float* src   = (const float*)d_in[2];   // [2,1,128,128]
  const float* mask  = (const float*)d_in[3];   // [2,1,128,128]

  float* outp   = (float*)d_out;
  float* y_pred = outp;                                   // 2*1024*1024
  float* cv     = outp + (size_t)2 * HH * WW;             // 2*1023*1024
  float* ch     = cv + (size_t)2 * (HH - 1) * WW;         // 2*1024*1023

  float* ws    = (float*)d_ws;
  float* shift = ws;          // 1 float
  float* bufA  = ws + 256;    // 2*1024*1024 floats (L2-resident ping buffer)

  const int npix = 2 * HH * WW;
  k_min<<<1, 256, 0, stream>>>(src, shift);
  k_cond<<<npix / 256, 256, 0, stream>>>(guide, ybic, cv, ch);
  k_init<<<npix / 256, 256, 0, stream>>>(ybic, shift, bufA);

  dim3 grid(WW / 64, HH / 8, 2);
  float* cur = bufA;
  float* nxt = y_pred;        // ping-pong with the output region (all in L2)
  for (int it = 0; it < NPRE; ++it) {
    k_step<<<grid, 256, 0, stream>>>(cur, nxt, cv, ch, src, mask, shift);
    float* tmp = cur; cur = nxt; nxt = tmp;
  }
  // NPRE is even -> result sits in bufA; subtract shift into y_pred.
  k_final<<<npix / 256, 256, 0, stream>>>(cur, shift, y_pred);
}